// GCN_1382979469642
// MI455X (gfx1250) — compile-verified
//
#include <hip/hip_runtime.h>

typedef __attribute__((ext_vector_type(2))) float v2f;
typedef __attribute__((ext_vector_type(8))) float v8f;

#define N_NODES 100000
#define IN_FT   256
#define OUT_FT  64

// ---------------------------------------------------------------------------
// Stage 1: seq_fts[n][o] = sum_i seq[n][i] * W[o][i]   (fp32 WMMA 16x16x4)
// One wave -> 16 rows x 64 cols. 8 waves / 256-thread block -> 128 rows/block.
// ---------------------------------------------------------------------------
__global__ __launch_bounds__(256) void gcn_gemm_wmma(
    const float* __restrict__ seq, const float* __restrict__ W,
    float* __restrict__ seq_fts) {
  const int lane = threadIdx.x & 31;
  const int wave = threadIdx.x >> 5;
  const int row0 = (blockIdx.x * 8 + wave) * 16;
  if (row0 >= N_NODES) return;              // wave-uniform: EXEC stays all-1s

  const int l16  = lane & 15;
  const int half = lane >> 4;

  // A fragment source: lane m(+16) holds seq[row0+m][k + 2*half + {0,1}]
  const float* aptr = seq + (size_t)(row0 + l16) * IN_FT + 2 * half;
  // B fragment source: lane n(+16) holds W[col0+n][k + 2*half + {0,1}]
  const float* bptr = W + (size_t)l16 * IN_FT + 2 * half;

  v8f acc0 = {}, acc1 = {}, acc2 = {}, acc3 = {};

#pragma unroll 4
  for (int k = 0; k < IN_FT; k += 4) {
    v2f a  = *(const v2f*)(aptr + k);
    v2f b0 = *(const v2f*)(bptr + 0 * 16 * IN_FT + k);
    v2f b1 = *(const v2f*)(bptr + 1 * 16 * IN_FT + k);
    v2f b2 = *(const v2f*)(bptr + 2 * 16 * IN_FT + k);
    v2f b3 = *(const v2f*)(bptr + 3 * 16 * IN_FT + k);
    acc0 = __builtin_amdgcn_wmma_f32_16x16x4_f32(false, a, false, b0, (short)0, acc0, false, false);
    acc1 = __builtin_amdgcn_wmma_f32_16x16x4_f32(false, a, false, b1, (short)0, acc1, false, false);
    acc2 = __builtin_amdgcn_wmma_f32_16x16x4_f32(false, a, false, b2, (short)0, acc2, false, false);
    acc3 = __builtin_amdgcn_wmma_f32_16x16x4_f32(false, a, false, b3, (short)0, acc3, false, false);
  }

  // D layout: VGPR r, lane l -> D[r + 8*(l>>4)][l&15]
  float* o = seq_fts + (size_t)(row0 + 8 * half) * OUT_FT + l16;
#pragma unroll
  for (int r = 0; r < 8; ++r) {
    o[(size_t)r * OUT_FT +  0] = acc0[r];
    o[(size_t)r * OUT_FT + 16] = acc1[r];
    o[(size_t)r * OUT_FT + 32] = acc2[r];
    o[(size_t)r * OUT_FT + 48] = acc3[r];
  }
}

// ---------------------------------------------------------------------------
// Stage 2a: out[n][f] = bias[f]   (d_out is poisoned; bias is the accumulator
// base for the segment-sum)
// ---------------------------------------------------------------------------
__global__ void gcn_init_out(float* __restrict__ out,
                             const float* __restrict__ bias, int n) {
  int i = blockIdx.x * blockDim.x + threadIdx.x;
  if (i < n) out[i] = bias[i & (OUT_FT - 1)];
}

// ---------------------------------------------------------------------------
// Stage 2b: scatter-add messages. Thread-per-(edge,feature); a wave covers
// 32 consecutive features of one edge -> coalesced gather, uniform indices.
// seq_fts (25.6MB) and out stay L2-resident (192MB L2), atomics hit L2.
// ---------------------------------------------------------------------------
__global__ __launch_bounds__(256) void gcn_spmm_scatter(
    const int* __restrict__ ei, const float* __restrict__ ev,
    const float* __restrict__ seq_fts, float* __restrict__ out, int E) {
  unsigned gid = blockIdx.x * 256u + threadIdx.x;
  int e = (int)(gid >> 6);
  int f = (int)(gid & 63u);
  if (e >= E) return;
  int dst = ei[e];       // edge_index[0][e]
  int src = ei[E + e];   // edge_index[1][e]
  float v = ev[e] * seq_fts[(size_t)src * OUT_FT + f];
  unsafeAtomicAdd(&out[(size_t)dst * OUT_FT + f], v);
}

// ---------------------------------------------------------------------------
// Stage 3: PReLU with shared alpha
// ---------------------------------------------------------------------------
__global__ void gcn_prelu(float* __restrict__ out,
                          const float* __restrict__ alpha_p, int n) {
  int i = blockIdx.x * blockDim.x + threadIdx.x;
  if (i < n) {
    float x = out[i];
    float a = *alpha_p;
    out[i] = (x >= 0.0f) ? x : a * x;
  }
}

extern "C" void kernel_launch(void* const* d_in, const int* in_sizes, int n_in,
                              void* d_out, int out_size, void* d_ws, size_t ws_size,
                              hipStream_t stream) {
  const float* seq   = (const float*)d_in[0];   // [N, 256]
  const int*   ei    = (const int*)  d_in[1];   // [2, E] row0=dst, row1=src
  const float* ev    = (const float*)d_in[2];   // [E]
  const float* W     = (const float*)d_in[3];   // [64, 256]
  const float* bias  = (const float*)d_in[4];   // [64]
  const float* alpha = (const float*)d_in[5];   // [1]
  float* out = (float*)d_out;
  float* seq_fts = (float*)d_ws;                // [N, 64] = 25.6 MB scratch
  const int E = in_sizes[2];

  const int n_tiles = N_NODES / 16;             // 6250, exact
  gcn_gemm_wmma<<<(n_tiles + 7) / 8, 256, 0, stream>>>(seq, W, seq_fts);

  const int n_out = N_NODES * OUT_FT;           // 6.4M
  gcn_init_out<<<(n_out + 255) / 256, 256, 0, stream>>>(out, bias, n_out);

  unsigned total = (unsigned)E * 64u;           // 102.4M threads
  gcn_spmm_scatter<<<(total + 255u) / 256u, 256, 0, stream>>>(ei, ev, seq_fts, out, E);

  gcn_prelu<<<(n_out + 255) / 256, 256, 0, stream>>>(out, alpha, n_out);
}